// GRUDecoder_54589034332755
// MI455X (gfx1250) — compile-verified
//
#include <hip/hip_runtime.h>
#include <hip/hip_bf16.h>
#include <math.h>

// ---------------------------------------------------------------------------
// GRU decoder step, algebraically reduced to matvecs (memory-bound, ~40us at
// 23.3 TB/s).  Row-major GEMVs run on the CDNA5 f32 matrix pipe via
// V_WMMA_F32_16X16X4_F32 (one wave32 = 16 output rows; A = 16x4 f32 slab,
// one b64 load/lane; B = input vector in column N=0).  Every GEMV is
// K-split into 4-8 chunks (wave = tile*nchunks + chunk) to raise the wave
// count 4-8x: a 23.3 TB/s part needs MBs of loads in flight, and the
// un-split versions (64-512 waves) cannot provide that.  Reductions are
// deterministic fixed-tree sums; one shared partial region serves all
// (strictly stream-sequential) stages.
// ---------------------------------------------------------------------------

typedef __attribute__((ext_vector_type(2))) float v2f;
typedef __attribute__((ext_vector_type(8))) float v8f;

#define Hh    2048
#define Vv    32
#define Ll    5
#define Tt    8192
#define H2    (2 * Hh)     // 4096
#define H3    (3 * Hh)     // 6144
#define Hhalf (Hh / 2)     // 1024

// ---- workspace layout (float offsets) ----
#define OFF_Q     0u        // 2048   q = Wq@s + bq
#define OFF_W     2048u     // 4096   w = Wk^T q
#define OFF_ATT   6144u     // 8192   attention logits
#define OFF_P     14336u    // 8192   exp(att - max)
#define OFF_U     22528u    // 4096   u = enc^T p (unnormalized)
#define OFF_GIN   26624u    // 4096   gru_input = [tok ; c]
#define OFF_HRAW  30720u    // 10240  raw h_new per layer
#define OFF_GI    40960u    // 6144   gi gates
#define OFF_GH    47104u    // 6144   gh gates
#define OFF_PLN   53248u    // 6144   layernormed pre
#define OFF_Z1    59392u    // 1024   gelu(W1@pre)
#define OFF_RED   60416u    // 16     [0]=sumExp
#define OFF_GPART 60432u    // 131072 shared partial region (max user: 32x4096)

// ---------------------------------------------------------------------------
// WMMA f32 16x16x4 GEMV tile over k in [k0, k0+kc).
// A layout (32b 16x4): lane<16 -> {A[m,k],A[m,k+1]}, lane>=16 -> {A[m,k+2],
// A[m,k+3]}, m = lane&15.  B mirrors it; only N==0 lanes (0,16) carry x.
// ---------------------------------------------------------------------------
__device__ __forceinline__ v8f wmma_tile(const float* __restrict__ W, int ld,
                                         const float* __restrict__ x,
                                         int r0, int k0, int kc) {
  const int lane = threadIdx.x & 31;
  const int m    = lane & 15;
  const int kh   = (lane >> 4) << 1;        // 0 or 2
  const bool n0  = (m == 0);
  const float* wrow = W + (size_t)(r0 + m) * (size_t)ld + kh + k0;
  const float* xk   = x + kh + k0;
  const v2f z2 = {0.f, 0.f};
  v8f a0 = {0.f,0.f,0.f,0.f,0.f,0.f,0.f,0.f};
  v8f a1 = {0.f,0.f,0.f,0.f,0.f,0.f,0.f,0.f};
  for (int k = 0; k < kc; k += 8) {
    v2f w0 = *(const v2f*)(wrow + k);
    v2f w1 = *(const v2f*)(wrow + k + 4);
    v2f x0 = *(const v2f*)(xk + k);
    v2f x1 = *(const v2f*)(xk + k + 4);
    v2f b0 = n0 ? x0 : z2;
    v2f b1 = n0 ? x1 : z2;
    a0 = __builtin_amdgcn_wmma_f32_16x16x4_f32(false, w0, false, b0,
                                               (short)0, a0, false, false);
    a1 = __builtin_amdgcn_wmma_f32_16x16x4_f32(false, w1, false, b1,
                                               (short)0, a1, false, false);
  }
  return a0 + a1;
}

__device__ __forceinline__ float gelu_tanh(float t) {
  return 0.5f * t * (1.f + tanhf(0.79788456080286536f * (t + 0.044715f * t * t * t)));
}

// C/D layout: lane 0 VGPR v = D[v,0]; lane 16 VGPR v = D[8+v,0].
__device__ __forceinline__ void wmma_store16(float* __restrict__ y,
                                             const float* __restrict__ bias,
                                             const float* __restrict__ add,
                                             float scale, int r0, v8f acc, int act) {
  const int lane = threadIdx.x & 31;
  if ((lane & 15) == 0) {
    int base = r0 + ((lane >> 4) << 3);
#pragma unroll
    for (int i = 0; i < 8; ++i) {
      int r = base + i;
      float v = acc[i] * scale;
      if (bias) v += bias[r];
      if (add)  v += add[r];
      if (act)  v = gelu_tanh(v);
      y[r] = v;
    }
  }
}

// Direct (un-split) GEMV: used only for the tiny logits matvec.
__global__ void wmma_gemv(const float* __restrict__ W, const float* __restrict__ x,
                          const float* __restrict__ bias, const float* __restrict__ add,
                          const float* __restrict__ sum_ptr, float* __restrict__ y,
                          int C, int R, int act) {
  int wave = (int)((blockIdx.x * blockDim.x + threadIdx.x) >> 5);
  int r0 = wave * 16;
  if (r0 >= R) return;                       // wave-uniform: EXEC stays all-1s
  v8f acc = wmma_tile(W, C, x, r0, 0, C);
  float scale = sum_ptr ? (1.0f / sum_ptr[0]) : 1.0f;
  wmma_store16(y, bias, add, scale, r0, acc, act);
}

// K-split GEMV: wave = tile*(1<<lg) + chunk; partial[chunk*R + r] written.
__global__ void wmma_gemv_part(const float* __restrict__ W, const float* __restrict__ x,
                               float* __restrict__ part, int C, int R, int lg) {
  int wave = (int)((blockIdx.x * blockDim.x + threadIdx.x) >> 5);
  int n     = 1 << lg;
  int tile  = wave >> lg;
  int chunk = wave & (n - 1);
  int r0 = tile * 16;
  if (r0 >= R) return;                       // wave-uniform
  int Kc = C >> lg;
  v8f acc = wmma_tile(W, C, x, r0, chunk * Kc, Kc);
  wmma_store16(part + (size_t)chunk * R, nullptr, nullptr, 1.0f, r0, acc, 0);
}

// y[r] = (sum_c part[c*R+r]) * scale + bias + add, optional gelu. Deterministic.
__global__ void gemv_reduce(const float* __restrict__ part, int nchunks, int R,
                            const float* __restrict__ bias, const float* __restrict__ add,
                            const float* __restrict__ sum_ptr, float* __restrict__ y,
                            int act) {
  int r = blockIdx.x * blockDim.x + threadIdx.x;
  if (r >= R) return;
  float s = 0.f;
  for (int c = 0; c < nchunks; ++c) s += part[(size_t)c * R + r];
  float scale = sum_ptr ? (1.0f / sum_ptr[0]) : 1.0f;
  s *= scale;
  if (bias) s += bias[r];
  if (add)  s += add[r];
  if (act)  s = gelu_tanh(s);
  y[r] = s;
}

// Transposed matvec, row-chunked: part[ry*cols + j] = sum_{i in chunk} W[i,j]*v[i]
__global__ void colmatvec_part(const float* __restrict__ W, const float* __restrict__ v,
                               float* __restrict__ part, int cols, int rchunk) {
  int j  = blockIdx.x * blockDim.x + threadIdx.x;
  int i0 = blockIdx.y * rchunk;
  float s = 0.f;
  for (int i = i0; i < i0 + rchunk; ++i) s = fmaf(W[(size_t)i * cols + j], v[i], s);
  part[(size_t)blockIdx.y * cols + j] = s;
}

// single block: token-embedding copy + softmax max / exp / sum (bk*q constant
// dropped: softmax-invariant).
__global__ void softmax_prep(const float* __restrict__ att, float* __restrict__ p,
                             float* __restrict__ red, const float* __restrict__ emb,
                             const int* __restrict__ shot, float* __restrict__ gin) {
  __shared__ float sd[1024];
  int tid = threadIdx.x;
  int tok = shot[0];
  for (int j = tid; j < Hh; j += 1024) gin[j] = emb[(size_t)tok * Hh + j];
  float m = -INFINITY;
  for (int t = tid; t < Tt; t += 1024) m = fmaxf(m, att[t]);
  sd[tid] = m; __syncthreads();
  for (int s = 512; s > 0; s >>= 1) { if (tid < s) sd[tid] = fmaxf(sd[tid], sd[tid + s]); __syncthreads(); }
  float mx = sd[0]; __syncthreads();
  float sum = 0.f;
  for (int t = tid; t < Tt; t += 1024) { float e = expf(att[t] - mx); p[t] = e; sum += e; }
  sd[tid] = sum; __syncthreads();
  for (int s = 512; s > 0; s >>= 1) { if (tid < s) sd[tid] += sd[tid + s]; __syncthreads(); }
  if (tid == 0) red[0] = sd[0];
}

// part[chunk*H2 + j] = sum_{t in chunk} p[t] * enc[t, j]   (deterministic)
__global__ void u_accum(const float* __restrict__ enc, const float* __restrict__ p,
                        float* __restrict__ part) {
  int j  = blockIdx.x * blockDim.x + threadIdx.x;  // 0..4095
  int t0 = blockIdx.y * 256;
  float s = 0.f;
  for (int t = t0; t < t0 + 256; ++t) s = fmaf(p[t], enc[(size_t)t * H2 + j], s);
  part[(size_t)blockIdx.y * H2 + j] = s;
}

__global__ void gru_combine(const float* __restrict__ gi, const float* __restrict__ gh,
                            const float* __restrict__ hprev, float* __restrict__ hnew) {
  int j = blockIdx.x * blockDim.x + threadIdx.x;
  if (j >= Hh) return;
  float r = 1.f / (1.f + expf(-(gi[j] + gh[j])));
  float z = 1.f / (1.f + expf(-(gi[j + Hh] + gh[j + Hh])));
  float n = tanhf(gi[j + 2 * Hh] + r * gh[j + 2 * Hh]);
  hnew[j] = (1.f - z) * n + z * hprev[j];
}

// per-layer layernorm of raw h -> d_out[32 + l*H ..]
__global__ void ln_h(const float* __restrict__ hraw, const float* __restrict__ g,
                     const float* __restrict__ b, float* __restrict__ out) {
  __shared__ float sd[256];
  int l = blockIdx.x, tid = threadIdx.x;
  const float* h = hraw + (size_t)l * Hh;
  float s = 0.f;
  for (int j = tid; j < Hh; j += 256) s += h[j];
  sd[tid] = s; __syncthreads();
  for (int k = 128; k > 0; k >>= 1) { if (tid < k) sd[tid] += sd[tid + k]; __syncthreads(); }
  float mean = sd[0] / (float)Hh; __syncthreads();
  float v = 0.f;
  for (int j = tid; j < Hh; j += 256) { float d = h[j] - mean; v += d * d; }
  sd[tid] = v; __syncthreads();
  for (int k = 128; k > 0; k >>= 1) { if (tid < k) sd[tid] += sd[tid + k]; __syncthreads(); }
  float inv = rsqrtf(sd[0] / (float)Hh + 1e-5f); __syncthreads();
  for (int j = tid; j < Hh; j += 256)
    out[(size_t)l * Hh + j] = (h[j] - mean) * inv * g[j] + b[j];
}

// pre = [x_last(H) ; gru_input(2H)], layernorm over 3H -> preln
__global__ void ln_pre(const float* __restrict__ xlast, const float* __restrict__ gin,
                       const float* __restrict__ g, const float* __restrict__ b,
                       float* __restrict__ out) {
  __shared__ float sd[1024];
  int tid = threadIdx.x;
  float s = 0.f;
  for (int i = tid; i < H3; i += 1024) {
    float v = (i < Hh) ? xlast[i] : gin[i - Hh];
    s += v;
  }
  sd[tid] = s; __syncthreads();
  for (int k = 512; k > 0; k >>= 1) { if (tid < k) sd[tid] += sd[tid + k]; __syncthreads(); }
  float mean = sd[0] / (float)H3; __syncthreads();
  float v = 0.f;
  for (int i = tid; i < H3; i += 1024) {
    float x = ((i < Hh) ? xlast[i] : gin[i - Hh]) - mean;
    v += x * x;
  }
  sd[tid] = v; __syncthreads();
  for (int k = 512; k > 0; k >>= 1) { if (tid < k) sd[tid] += sd[tid + k]; __syncthreads(); }
  float inv = rsqrtf(sd[0] / (float)H3 + 1e-5f); __syncthreads();
  for (int i = tid; i < H3; i += 1024) {
    float x = (i < Hh) ? xlast[i] : gin[i - Hh];
    out[i] = (x - mean) * inv * g[i] + b[i];
  }
}

extern "C" void kernel_launch(void* const* d_in, const int* in_sizes, int n_in,
                              void* d_out, int out_size, void* d_ws, size_t ws_size,
                              hipStream_t stream) {
  const float* enc     = (const float*)d_in[0];
  const int*   shot    = (const int*)d_in[1];
  const float* hidden  = (const float*)d_in[2];
  const float* emb     = (const float*)d_in[3];
  const float* Wq      = (const float*)d_in[4];
  const float* bq      = (const float*)d_in[5];
  const float* Wk      = (const float*)d_in[6];
  /* bk (d_in[7]) dropped: softmax-invariant constant */
  const float* Wv      = (const float*)d_in[8];
  const float* bv      = (const float*)d_in[9];
  const float* W_ih0   = (const float*)d_in[10];
  const float* W_ihr   = (const float*)d_in[11];
  const float* W_hh    = (const float*)d_in[12];
  const float* b_ih    = (const float*)d_in[13];
  const float* b_hh    = (const float*)d_in[14];
  const float* g_hid   = (const float*)d_in[15];
  const float* be_hid  = (const float*)d_in[16];
  const float* g_out   = (const float*)d_in[17];
  const float* be_out  = (const float*)d_in[18];
  const float* W1      = (const float*)d_in[19];
  const float* b1      = (const float*)d_in[20];
  const float* W2      = (const float*)d_in[21];
  const float* b2      = (const float*)d_in[22];

  float* ws   = (float*)d_ws;
  float* q    = ws + OFF_Q;
  float* w    = ws + OFF_W;
  float* att  = ws + OFF_ATT;
  float* p    = ws + OFF_P;
  float* u    = ws + OFF_U;
  float* gin  = ws + OFF_GIN;
  float* hraw = ws + OFF_HRAW;
  float* gi   = ws + OFF_GI;
  float* gh   = ws + OFF_GH;
  float* pln  = ws + OFF_PLN;
  float* z1   = ws + OFF_Z1;
  float* red  = ws + OFF_RED;
  float* part = ws + OFF_GPART;   // shared, stages are stream-sequential

  float* logits = (float*)d_out;          // [32]
  float* h_out  = (float*)d_out + Vv;     // [5*2048]

  const float* s_t = hidden + (size_t)(Ll - 1) * Hh;

  // 1) q = Wq @ s_t + bq             (8-way K-split: 1024 waves)
  wmma_gemv_part<<<128, 256, 0, stream>>>(Wq, s_t, part, Hh, Hh, 3);
  gemv_reduce<<<Hh / 256, 256, 0, stream>>>(part, 8, Hh, bq, nullptr, nullptr, q, 0);
  // 2) w = Wk^T q                    (16 row-chunks x 4096 cols: 256 blocks)
  colmatvec_part<<<dim3(H2 / 256, 16), 256, 0, stream>>>(Wk, q, part, H2, Hh / 16);
  gemv_reduce<<<H2 / 256, 256, 0, stream>>>(part, 16, H2, nullptr, nullptr, nullptr, w, 0);
  // 3) att = enc @ w                 (4-way K-split over 128 MB: 2048 waves)
  wmma_gemv_part<<<256, 256, 0, stream>>>(enc, w, part, H2, Tt, 2);
  gemv_reduce<<<Tt / 256, 256, 0, stream>>>(part, 4, Tt, nullptr, nullptr, nullptr, att, 0);
  // 4) softmax prep + token embedding copy
  softmax_prep<<<1, 1024, 0, stream>>>(att, p, red, emb, shot, gin);
  // 5) u = enc^T p                   (32 t-chunks, deterministic)
  u_accum<<<dim3(H2 / 256, Tt / 256), 256, 0, stream>>>(enc, p, part);
  gemv_reduce<<<H2 / 256, 256, 0, stream>>>(part, 32, H2, nullptr, nullptr, nullptr, u, 0);
  // 6) c = Wv @ (u / sumExp) + bv + s_t -> gru_input[H:2H]   (8-way split)
  wmma_gemv_part<<<128, 256, 0, stream>>>(Wv, u, part, H2, Hh, 3);
  gemv_reduce<<<Hh / 256, 256, 0, stream>>>(part, 8, Hh, bv, s_t, red, gin + Hh, 0);
  // 7) 5-layer GRU step              (4-way split: 1536 waves per matvec)
  for (int l = 0; l < Ll; ++l) {
    const float* Wi  = (l == 0) ? W_ih0 : (W_ihr + (size_t)(l - 1) * H3 * Hh);
    int Ci           = (l == 0) ? H2 : Hh;
    const float* xin = (l == 0) ? gin : (hraw + (size_t)(l - 1) * Hh);
    wmma_gemv_part<<<192, 256, 0, stream>>>(Wi, xin, part, Ci, H3, 2);
    gemv_reduce<<<H3 / 256, 256, 0, stream>>>(part, 4, H3, b_ih + (size_t)l * H3,
                                              nullptr, nullptr, gi, 0);
    wmma_gemv_part<<<192, 256, 0, stream>>>(W_hh + (size_t)l * H3 * Hh,
                                            hidden + (size_t)l * Hh, part, Hh, H3, 2);
    gemv_reduce<<<H3 / 256, 256, 0, stream>>>(part, 4, H3, b_hh + (size_t)l * H3,
                                              nullptr, nullptr, gh, 0);
    gru_combine<<<Hh / 256, 256, 0, stream>>>(gi, gh, hidden + (size_t)l * Hh,
                                              hraw + (size_t)l * Hh);
  }
  // 8) h output = layernorm(h_raw)
  ln_h<<<Ll, 256, 0, stream>>>(hraw, g_hid, be_hid, h_out);
  // 9) pre = LN([x_last ; gru_input])
  ln_pre<<<1, 1024, 0, stream>>>(hraw + (size_t)(Ll - 1) * Hh, gin, g_out, be_out, pln);
  // 10) z1 = gelu(W1 @ pre + b1)     (8-way split: 512 waves)
  wmma_gemv_part<<<64, 256, 0, stream>>>(W1, pln, part, H3, Hhalf, 3);
  gemv_reduce<<<Hhalf / 256, 256, 0, stream>>>(part, 8, Hhalf, b1, nullptr, nullptr, z1, 1);
  // 11) logits = W2 @ z1 + b2        (direct WMMA, 2 waves)
  wmma_gemv<<<1, 256, 0, stream>>>(W2, z1, b2, nullptr, nullptr, logits, Hhalf, Vv, 0);
}